// MM_77395310674147
// MI455X (gfx1250) — compile-verified
//
#include <hip/hip_runtime.h>
#include <hip/hip_bf16.h>
#include <math.h>

// ---------------------------------------------------------------------------
// Problem constants (match reference)
// ---------------------------------------------------------------------------
static constexpr int B_   = 8;
static constexpr int HQ_  = 32;
static constexpr int HKV_ = 8;
static constexpr int D_   = 128;
static constexpr int G_   = HQ_ / HKV_;          // 4
static constexpr int BS_  = 128;                 // page size (tokens)
static constexpr int MAX_CMP_BLOCKS_ = 16;
static constexpr int MAX_ORI_BLOCKS_ = 64;
static constexpr int TOK_STRIDE_ = HKV_ * D_;    // 1024 floats between tokens
static constexpr int PAGE_STRIDE_ = 2 * BS_ * HKV_ * D_;  // 262144 floats / page
static constexpr int KV_PLANE_ = BS_ * HKV_ * D_;         // 131072 floats K->V
static constexpr float SCALE_ = 0.08838834764831845f;     // 1/sqrt(128)

typedef __attribute__((ext_vector_type(2))) float v2f;
typedef __attribute__((ext_vector_type(8))) float v8f;

// ---------------------------------------------------------------------------
// Kernel 1: per (batch*kv_head, cmp_page) flash-attention partial using
// V_WMMA_F32_16X16X4_F32.  One wave32 per block.
//
// Score tile:  S^T = K_tile(16 tok x 4 dch) * Q^T(4 dch x 16 heads)
//   A lane L<16 : K[tok=L][k0+0..1] ;  L>=16 : K[tok=L-16][k0+2..3]
//   B lane n<16 : Q[head n][k0+0..1];  n>=16 : Q[head n-16][k0+2..3]
//   D VGPR j, lane n<16 : S^T[tok=j][head n]; lane n>=16: S^T[tok=j+8][head n-16]
// PV tile:     O^T += V^T(16 d x 4 tok) * P^T(4 tok x 16 heads)
//
// The score reduction over D=128 (32 WMMAs) uses 4 independent accumulators
// to break the C-operand RAW chain in the matrix pipe.
// ---------------------------------------------------------------------------
__global__ __launch_bounds__(32)
void attn_cmp_pages_kernel(const float* __restrict__ q,
                           const float* __restrict__ cmp_kv,
                           const int*   __restrict__ cmp_block_table,
                           const int*   __restrict__ seqused,
                           float* __restrict__ ws_acc,   // [64][16][G][D]
                           float* __restrict__ ws_m,     // [64][16][G]
                           float* __restrict__ ws_l)     // [64][16][G]
{
    const int pair  = blockIdx.x;          // b*HKV + kv
    const int b     = pair >> 3;
    const int kv    = pair & 7;
    const int chunk = blockIdx.y;          // compressed page index
    const int lane  = threadIdx.x;

    const int seq     = seqused[b];
    const int cmp_len = seq >> 2;          // seq / RATIO
    const int tok0    = chunk * BS_;
    if (tok0 >= cmp_len) return;           // uniform per block

    const int page = cmp_block_table[b * MAX_CMP_BLOCKS_ + chunk];
    const float* Kbase = cmp_kv + (size_t)page * PAGE_STRIDE_ + (size_t)kv * D_;
    const float* Vbase = Kbase + KV_PLANE_;

    const int hi  = lane >> 4;             // lane half
    const int l15 = lane & 15;

    // Q in WMMA-B layout, kept resident (32 x v2f = 64 VGPRs).
    const float* Qb = q + ((size_t)(b * HQ_ + kv * G_ + (l15 & 3))) * D_ + 2 * hi;
    v2f bq[32];
#pragma unroll
    for (int c = 0; c < 32; ++c)
        bq[c] = *(const v2f*)(Qb + 4 * c);

    float m_run = -INFINITY;
    float l_run = 0.0f;
    v8f acc[8];
#pragma unroll
    for (int dd = 0; dd < 8; ++dd) acc[dd] = {};

    const bool lo = (hi == 0);

    for (int tile = 0; tile < 8; ++tile) {
        const int tbase = tok0 + tile * 16;
        if (tbase >= cmp_len) break;

        // ---- scores: 32 fp32 WMMAs over D=128, 4-way ILP ----------------
        const float* Krow = Kbase + (size_t)(tile * 16 + l15) * TOK_STRIDE_ + 2 * hi;
        v8f sa0 = {}, sa1 = {}, sa2 = {}, sa3 = {};
#pragma unroll
        for (int c = 0; c < 32; c += 4) {
            v2f a0 = *(const v2f*)(Krow + 4 * (c + 0));
            v2f a1 = *(const v2f*)(Krow + 4 * (c + 1));
            v2f a2 = *(const v2f*)(Krow + 4 * (c + 2));
            v2f a3 = *(const v2f*)(Krow + 4 * (c + 3));
            sa0 = __builtin_amdgcn_wmma_f32_16x16x4_f32(
                      false, a0, false, bq[c + 0], (short)0, sa0, false, false);
            sa1 = __builtin_amdgcn_wmma_f32_16x16x4_f32(
                      false, a1, false, bq[c + 1], (short)0, sa1, false, false);
            sa2 = __builtin_amdgcn_wmma_f32_16x16x4_f32(
                      false, a2, false, bq[c + 2], (short)0, sa2, false, false);
            sa3 = __builtin_amdgcn_wmma_f32_16x16x4_f32(
                      false, a3, false, bq[c + 3], (short)0, sa3, false, false);
        }

        // ---- mask + online softmax --------------------------------------
        float sv[8];
        float mt = -INFINITY;
#pragma unroll
        for (int j = 0; j < 8; ++j) {
            const float sj = (sa0[j] + sa1[j]) + (sa2[j] + sa3[j]);
            const int tok = tbase + hi * 8 + j;
            const float x = (tok < cmp_len) ? sj * SCALE_ : -1e30f;
            sv[j] = x;
            mt = fmaxf(mt, x);
        }
        mt = fmaxf(mt, __shfl_xor(mt, 16, 32));        // per-head max (16 tok)
        const float m_new = fmaxf(m_run, mt);
        const float corr  = __expf(m_run - m_new);     // 0 on first tile
        m_run = m_new;

        float p[8];
        float ps = 0.0f;
#pragma unroll
        for (int j = 0; j < 8; ++j) {
            p[j] = __expf(sv[j] - m_new);              // underflows to 0 if masked
            ps += p[j];
        }
        l_run = l_run * corr + ps;

#pragma unroll
        for (int dd = 0; dd < 8; ++dd)
#pragma unroll
            for (int j = 0; j < 8; ++j)
                acc[dd][j] *= corr;

        // ---- PV: 4 token-groups x 8 d-chunks of WMMA --------------------
#pragma unroll
        for (int tq = 0; tq < 4; ++tq) {
            const int pb = (tq & 1) * 4;
            // cross-half exchange to build P^T B-operand
            const float s0 = __shfl_xor(p[pb + 0], 16, 32);
            const float s1 = __shfl_xor(p[pb + 1], 16, 32);
            const float s2 = __shfl_xor(p[pb + 2], 16, 32);
            const float s3 = __shfl_xor(p[pb + 3], 16, 32);
            v2f bp;
            if (tq < 2) { bp.x = lo ? p[pb + 0] : s2;  bp.y = lo ? p[pb + 1] : s3; }
            else        { bp.x = lo ? s0 : p[pb + 2];  bp.y = lo ? s1 : p[pb + 3]; }

            const float* Vt = Vbase
                + (size_t)(tile * 16 + tq * 4 + 2 * hi) * TOK_STRIDE_ + l15;
#pragma unroll
            for (int dd = 0; dd < 8; ++dd) {
                v2f av;
                av.x = Vt[dd * 16];                  // token t0+2*hi
                av.y = Vt[dd * 16 + TOK_STRIDE_];    // token t0+2*hi+1
                acc[dd] = __builtin_amdgcn_wmma_f32_16x16x4_f32(
                              false, av, false, bp, (short)0, acc[dd], false, false);
            }
        }
    }

    // ---- store per-page partial ----------------------------------------
    const float l_tot = l_run + __shfl_xor(l_run, 16, 32);
    const size_t pc = (size_t)pair * MAX_CMP_BLOCKS_ + chunk;
    if (l15 < G_) {
        if (hi == 0) {
            ws_m[pc * G_ + l15] = m_run;
            ws_l[pc * G_ + l15] = l_tot;
        }
        float* ap = ws_acc + (pc * G_ + l15) * D_ + hi * 8;
#pragma unroll
        for (int dd = 0; dd < 8; ++dd)
#pragma unroll
            for (int j = 0; j < 8; ++j)
                ap[dd * 16 + j] = acc[dd][j];
    }
}

// ---------------------------------------------------------------------------
// Kernel 2: merge page partials, add residual tail (<=3 tokens) from ori
// cache, add sink, normalize.  One wave32 per (b, kv_head).
// Lane t owns d = t + 32*i, i in [0,4).
// ---------------------------------------------------------------------------
__global__ __launch_bounds__(32)
void attn_reduce_kernel(const float* __restrict__ q,
                        const float* __restrict__ ori_kv,
                        const float* __restrict__ sinks,
                        const int*   __restrict__ ori_block_table,
                        const int*   __restrict__ seqused,
                        const float* __restrict__ ws_acc,
                        const float* __restrict__ ws_m,
                        const float* __restrict__ ws_l,
                        float* __restrict__ out)
{
    const int pair = blockIdx.x;
    const int b    = pair >> 3;
    const int kv   = pair & 7;
    const int t    = threadIdx.x;

    const int seq      = seqused[b];
    const int cmp_len  = seq >> 2;
    const int cmp_base = cmp_len << 2;
    const int resid    = seq - cmp_base;     // 0..3

    float qreg[G_][4];
#pragma unroll
    for (int g = 0; g < G_; ++g)
#pragma unroll
        for (int i = 0; i < 4; ++i)
            qreg[g][i] = q[((size_t)(b * HQ_ + kv * G_ + g)) * D_ + t + 32 * i];

    float m[G_], l[G_], acc[G_][4];
#pragma unroll
    for (int g = 0; g < G_; ++g) {
        m[g] = -INFINITY;
        l[g] = 0.0f;
#pragma unroll
        for (int i = 0; i < 4; ++i) acc[g][i] = 0.0f;
    }

    // ---- merge compressed-page partials --------------------------------
    for (int c = 0; c < MAX_CMP_BLOCKS_; ++c) {
        if (c * BS_ >= cmp_len) break;
        const size_t pc = (size_t)pair * MAX_CMP_BLOCKS_ + c;
#pragma unroll
        for (int g = 0; g < G_; ++g) {
            const float mc = ws_m[pc * G_ + g];
            const float lc = ws_l[pc * G_ + g];
            const float mn = fmaxf(m[g], mc);
            const float a  = __expf(m[g] - mn);
            const float bf = __expf(mc   - mn);
            l[g] = l[g] * a + lc * bf;
            const float* ap = ws_acc + (pc * G_ + g) * D_;
#pragma unroll
            for (int i = 0; i < 4; ++i)
                acc[g][i] = acc[g][i] * a + ap[t + 32 * i] * bf;
            m[g] = mn;
        }
    }

    // ---- residual tail tokens from the ori cache -----------------------
    for (int j = 0; j < resid; ++j) {
        const int tok  = cmp_base + j;
        const int page = ori_block_table[b * MAX_ORI_BLOCKS_ + (tok >> 7)];
        const float* kp = ori_kv + (size_t)page * PAGE_STRIDE_
                        + (size_t)(tok & (BS_ - 1)) * TOK_STRIDE_ + kv * D_;
        const float* vp = kp + KV_PLANE_;
        float kr[4], vr[4];
#pragma unroll
        for (int i = 0; i < 4; ++i) {
            kr[i] = kp[t + 32 * i];
            vr[i] = vp[t + 32 * i];
        }
#pragma unroll
        for (int g = 0; g < G_; ++g) {
            float dot = 0.0f;
#pragma unroll
            for (int i = 0; i < 4; ++i) dot += qreg[g][i] * kr[i];
#pragma unroll
            for (int o = 16; o > 0; o >>= 1) dot += __shfl_xor(dot, o, 32);
            const float s  = dot * SCALE_;
            const float mn = fmaxf(m[g], s);
            const float a  = __expf(m[g] - mn);
            const float p  = __expf(s    - mn);
            l[g] = l[g] * a + p;
#pragma unroll
            for (int i = 0; i < 4; ++i)
                acc[g][i] = acc[g][i] * a + p * vr[i];
            m[g] = mn;
        }
    }

    // ---- sink + normalize + store --------------------------------------
#pragma unroll
    for (int g = 0; g < G_; ++g) {
        const float s  = sinks[kv * G_ + g];
        const float mn = fmaxf(m[g], s);
        const float a  = __expf(m[g] - mn);
        l[g] = l[g] * a + __expf(s - mn);
        const float inv = 1.0f / l[g];
#pragma unroll
        for (int i = 0; i < 4; ++i)
            out[((size_t)(b * HQ_ + kv * G_ + g)) * D_ + t + 32 * i] =
                acc[g][i] * a * inv;
    }
}

// ---------------------------------------------------------------------------
// Host launcher
// ---------------------------------------------------------------------------
extern "C" void kernel_launch(void* const* d_in, const int* in_sizes, int n_in,
                              void* d_out, int out_size, void* d_ws, size_t ws_size,
                              hipStream_t stream) {
    const float* q         = (const float*)d_in[0];
    const float* cmp_kv    = (const float*)d_in[1];
    const float* sinks     = (const float*)d_in[2];
    const int*   cmp_tab   = (const int*)  d_in[3];
    const int*   seqused   = (const int*)  d_in[4];
    const float* ori_kv    = (const float*)d_in[5];
    const int*   ori_tab   = (const int*)  d_in[6];
    // d_in[7] = cmp_ratio (fixed at 4 per reference constants)
    float* out = (float*)d_out;

    const int NPAIR = B_ * HKV_;                      // 64
    float* ws_acc = (float*)d_ws;                     // [64][16][4][128]
    float* ws_m   = ws_acc + (size_t)NPAIR * MAX_CMP_BLOCKS_ * G_ * D_;
    float* ws_l   = ws_m   + (size_t)NPAIR * MAX_CMP_BLOCKS_ * G_;

    dim3 g1(NPAIR, MAX_CMP_BLOCKS_);
    attn_cmp_pages_kernel<<<g1, 32, 0, stream>>>(q, cmp_kv, cmp_tab, seqused,
                                                 ws_acc, ws_m, ws_l);
    attn_reduce_kernel<<<NPAIR, 32, 0, stream>>>(q, ori_kv, sinks, ori_tab,
                                                 seqused, ws_acc, ws_m, ws_l, out);
}